// SafetyLayer_47399259079459
// MI455X (gfx1250) — compile-verified
//
#include <hip/hip_runtime.h>

#define NA 20          // NUM_ACTIONS
#define SD 8           // STATE_DIM

// Clang-native 128-bit vector type (accepted by __builtin_nontemporal_*)
typedef float vf4 __attribute__((ext_vector_type(4)));

__global__ __launch_bounds__(256) void SafetyLayer_kernel(
    const float* __restrict__ logits,
    const float* __restrict__ state,
    float* __restrict__ out,
    int nrows)
{
    int r = blockIdx.x * blockDim.x + threadIdx.x;
    if (r >= nrows) return;

    // ---- Load one 20-float row as 5 x b128, non-temporal (streamed once) ----
    const vf4* lp = reinterpret_cast<const vf4*>(logits + (size_t)r * NA);
    vf4 v[5];
#pragma unroll
    for (int i = 0; i < 5; ++i) v[i] = __builtin_nontemporal_load(&lp[i]);

    float x[NA];
#pragma unroll
    for (int i = 0; i < 5; ++i) {
        x[4*i + 0] = v[i].x; x[4*i + 1] = v[i].y;
        x[4*i + 2] = v[i].z; x[4*i + 3] = v[i].w;
    }

    // ---- Stable softmax over 20 elements (all in VGPRs) ----
    float m = x[0];
#pragma unroll
    for (int i = 1; i < NA; ++i) m = fmaxf(m, x[i]);

    float e[NA];
    float S = 0.0f;
#pragma unroll
    for (int i = 0; i < NA; ++i) { e[i] = __expf(x[i] - m); S += e[i]; }
    float inv = 1.0f / S;

    // ---- Dangerous-combo penalties: p[a]*p[b] > 0.05 -> subtract 1.5 ----
    float p1  = e[1]  * inv, p2  = e[2]  * inv;
    float p3  = e[3]  * inv, p11 = e[11] * inv;
    float p17 = e[17] * inv, p18 = e[18] * inv;

    float pen0 = (p1  * p2  > 0.05f) ? 1.5f : 0.0f;
    float pen1 = (p3  * p11 > 0.05f) ? 1.5f : 0.0f;
    float pen2 = (p17 * p18 > 0.05f) ? 1.5f : 0.0f;

    x[1]  -= pen0;  x[2]  -= pen0;
    x[3]  -= pen1;  x[11] -= pen1;
    x[17] -= pen2;  x[18] -= pen2;

    // ---- Vitals adjustments (branchless selects) ----
    const float* st = state + (size_t)r * SD;
    float hr   = __builtin_nontemporal_load(&st[1]);
    float bp   = __builtin_nontemporal_load(&st[3]);
    float spo2 = __builtin_nontemporal_load(&st[5]);

    bool low_bp = bp < 85.0f;
    x[2] += low_bp          ? -5.0f : 0.0f;
    x[1] += low_bp          ?  0.5f : 0.0f;
    x[1] += (bp   > 160.0f) ? -3.0f : 0.0f;
    x[4] += (hr   > 130.0f) ?  0.3f : 0.0f;
    x[5] += (spo2 <  90.0f) ?  0.5f : 0.0f;

    // ---- Store row as 5 x b128, non-temporal ----
    vf4* op = reinterpret_cast<vf4*>(out + (size_t)r * NA);
#pragma unroll
    for (int i = 0; i < 5; ++i) {
        vf4 w;
        w.x = x[4*i + 0]; w.y = x[4*i + 1];
        w.z = x[4*i + 2]; w.w = x[4*i + 3];
        __builtin_nontemporal_store(w, &op[i]);
    }
}

extern "C" void kernel_launch(void* const* d_in, const int* in_sizes, int n_in,
                              void* d_out, int out_size, void* d_ws, size_t ws_size,
                              hipStream_t stream)
{
    const float* logits = (const float*)d_in[0];   // (B, 20) f32
    const float* state  = (const float*)d_in[1];   // (B, 8)  f32
    float*       out    = (float*)d_out;           // (B, 20) f32

    int nrows = in_sizes[0] / NA;
    const int block = 256;
    int grid = (nrows + block - 1) / block;

    SafetyLayer_kernel<<<grid, block, 0, stream>>>(logits, state, out, nrows);
}